// PairGeneration_25752623906845
// MI455X (gfx1250) — compile-verified
//
#include <hip/hip_runtime.h>
#include <hip/hip_bf16.h>

#define THREADS 256

// Native vector types (HIP's float4 is a class; builtins need real vectors).
typedef float v4f __attribute__((ext_vector_type(4)));
typedef int   v4i __attribute__((ext_vector_type(4)));

// AS-qualified pointee types for the gfx1250 async-to-LDS builtin
// (prototype per compiler diagnostic: (v4i AS1*, v4i AS3*, Imm, Imm)).
typedef __attribute__((address_space(1))) v4i g_v4i;   // global
typedef __attribute__((address_space(3))) v4i l_v4i;   // LDS

__device__ __forceinline__ void wait_async_zero() {
#if __has_builtin(__builtin_amdgcn_s_wait_asynccnt)
    __builtin_amdgcn_s_wait_asynccnt(0);
#else
    asm volatile("s_wait_asynccnt 0" ::: "memory");
#endif
}

__global__ __launch_bounds__(THREADS) void
PairGeneration_25752623906845_kernel(const float* __restrict__ x,
                                     float* __restrict__ out,
                                     int n, long long P) {
    extern __shared__ float xs[];   // n floats (32 KB for n=8192; WGP has 320 KB)
    const int tid = threadIdx.x;

    // ---- Stage x into LDS once per workgroup (async-tensor path, ASYNCcnt) ----
    float* xp = const_cast<float*>(x);
    const int n4 = n & ~3;
#if __has_builtin(__builtin_amdgcn_global_load_async_to_lds_b128)
    for (int k = tid * 4; k < n4; k += THREADS * 4) {
        __builtin_amdgcn_global_load_async_to_lds_b128(
            (g_v4i*)(xp + k), (l_v4i*)(xs + k), /*offset=*/0, /*cpol=*/0);
    }
    for (int k = n4 + tid; k < n; k += THREADS) xs[k] = x[k];  // tail (none for n=8192)
    wait_async_zero();
#else
    // Fallback: plain load + ds_store staging.
    for (int k = tid; k < n; k += THREADS) xs[k] = x[k];
#endif
    __syncthreads();

    // ---- Row worker: out1[base..base+L) = x[i]; out2[base..base+L) = x[i+1..] ----
    auto do_row = [&](int i) {
        const int L = n - 1 - i;
        if (L <= 0) return;
        const long long base = (long long)i * n - ((long long)i * (i + 1)) / 2;
        float* __restrict__ o1 = out + base;
        float* __restrict__ o2 = out + P + base;
        const float c = xs[i];
        const float* __restrict__ src = xs + i + 1;  // LDS reads (ds_load)

        // Scalar head until base is 16B aligned (o1/o2 share misalignment).
        const int mis  = (int)(base & 3);
        int head = mis ? (4 - mis) : 0;
        if (head > L) head = L;
        if (tid < head) { o1[tid] = c; o2[tid] = src[tid]; }

        // Vector body: float4 non-temporal streaming stores (bypass L2 rinse).
        const int nv = (L - head) >> 2;
        const v4f c4 = {c, c, c, c};
        v4f* __restrict__ o1v = (v4f*)(o1 + head);
        v4f* __restrict__ o2v = (v4f*)(o2 + head);
        for (int v = tid; v < nv; v += THREADS) {
            const int t = head + (v << 2);
            v4f w;
            w.x = src[t];     w.y = src[t + 1];
            w.z = src[t + 2]; w.w = src[t + 3];
            __builtin_nontemporal_store(c4, o1v + v);
            __builtin_nontemporal_store(w,  o2v + v);
        }

        // Scalar tail.
        const int done = head + (nv << 2);
        const int rem  = L - done;
        if (tid < rem) { o1[done + tid] = c; o2[done + tid] = src[done + tid]; }
    };

    // Balanced row pairing: block b handles rows (b, n-2-b); combined length ~= n-1.
    const int i1 = (int)blockIdx.x;
    const int i2 = (n - 2) - i1;
    do_row(i1);
    if (i2 > i1) do_row(i2);
}

extern "C" void kernel_launch(void* const* d_in, const int* in_sizes, int n_in,
                              void* d_out, int out_size, void* d_ws, size_t ws_size,
                              hipStream_t stream) {
    const float* x = (const float*)d_in[0];
    float* out = (float*)d_out;
    const int n = in_sizes[0];
    if (n < 2) return;
    const long long P = (long long)n * (n - 1) / 2;
    const int nblocks = n / 2;                      // 4096 for n=8192
    const size_t smem = (size_t)n * sizeof(float);  // 32 KB
    hipLaunchKernelGGL(PairGeneration_25752623906845_kernel,
                       dim3(nblocks), dim3(THREADS), smem, stream,
                       x, out, n, P);
}